// GraphEncoder_59322088292869
// MI455X (gfx1250) — compile-verified
//
#include <hip/hip_runtime.h>

#define IN_DIM  128
#define HEADS   8
#define HD      1024      // HEADS * IN_DIM
#define OUT_DIM 64
#define NPG     2048
#define NGRAPH  16
#define NTOK    128

typedef __attribute__((ext_vector_type(2))) float v2f;
typedef __attribute__((ext_vector_type(8))) float v8f;

// ---- monotonic float<->uint encoding for atomicMax-based segment-max ----
__device__ __forceinline__ unsigned fenc(float f) {
    unsigned u = __float_as_uint(f);
    return (u & 0x80000000u) ? ~u : (u | 0x80000000u);
}
__device__ __forceinline__ float fdec(unsigned u) {
    return (u & 0x80000000u) ? __uint_as_float(u & 0x7FFFFFFFu) : __uint_as_float(~u);
}

// ============================================================
// fp32 WMMA GEMM: out[M x ncols] = X[M x 128] @ W[128 x ncols] + bias (+ addend)
// Block = 128 threads = 4 waves. Each wave computes a 16(M) x 64(N) tile with
// 4 accumulators, reusing its A fragment across 4 B fragments (A-reuse 4x).
// Grid = (M/64, ncols/64); all shapes here divide exactly -> EXEC all-1s always.
// ============================================================
template <bool HAS_ADD>
__global__ void wmma_gemm128(const float* __restrict__ X, const float* __restrict__ W,
                             const float* __restrict__ bias, const float* __restrict__ addend,
                             float* __restrict__ out, int ncols)
{
    const int wave  = threadIdx.x >> 5;
    const int lane  = threadIdx.x & 31;
    const int mBase = (blockIdx.x << 6) + (wave << 4);
    const int nB    = blockIdx.y << 6;
    const int half  = lane >> 4;                 // 0: lanes 0-15 (K0/K1), 1: lanes 16-31 (K2/K3)
    const int l16   = lane & 15;
    const float* xrow = X + (size_t)(mBase + l16) * IN_DIM;

    v8f acc0 = {0.f,0.f,0.f,0.f,0.f,0.f,0.f,0.f};
    v8f acc1 = acc0, acc2 = acc0, acc3 = acc0;

    #pragma unroll 4
    for (int kk = 0; kk < IN_DIM; kk += 4) {
        const int k0 = kk + (half << 1);
        v2f A; A.x = xrow[k0]; A.y = xrow[k0 + 1];
        const float* w0 = W + (size_t)k0 * ncols + nB + l16;
        const float* w1 = w0 + ncols;
        v2f B0; B0.x = w0[0];  B0.y = w1[0];
        v2f B1; B1.x = w0[16]; B1.y = w1[16];
        v2f B2; B2.x = w0[32]; B2.y = w1[32];
        v2f B3; B3.x = w0[48]; B3.y = w1[48];
        acc0 = __builtin_amdgcn_wmma_f32_16x16x4_f32(false, A, false, B0, (short)0, acc0, false, false);
        acc1 = __builtin_amdgcn_wmma_f32_16x16x4_f32(false, A, false, B1, (short)0, acc1, false, false);
        acc2 = __builtin_amdgcn_wmma_f32_16x16x4_f32(false, A, false, B2, (short)0, acc2, false, false);
        acc3 = __builtin_amdgcn_wmma_f32_16x16x4_f32(false, A, false, B3, (short)0, acc3, false, false);
    }

    const float b0 = bias[nB + l16];
    const float b1 = bias[nB + 16 + l16];
    const float b2 = bias[nB + 32 + l16];
    const float b3 = bias[nB + 48 + l16];

    #pragma unroll
    for (int r = 0; r < 8; ++r) {
        const int row = mBase + r + (half << 3);       // C layout: M = r + 8*half
        const size_t o = (size_t)row * ncols + nB + l16;
        float v0 = acc0[r] + b0;
        float v1 = acc1[r] + b1;
        float v2 = acc2[r] + b2;
        float v3 = acc3[r] + b3;
        if constexpr (HAS_ADD) {
            const float* arow = addend + o;
            v0 += arow[0]; v1 += arow[16]; v2 += arow[32]; v3 += arow[48];
        }
        float* orow = out + o;
        orow[0] = v0; orow[16] = v1; orow[32] = v2; orow[48] = v3;
    }
}

// ============================================================
// Per-edge attention logits: alpha[e,h] = dot(Q[dst,h,:], K[src,h,:]) / sqrt(128)
// One wave per edge; float4 loads, wave32 shuffle reduction; segment-max via atomicMax.
// Also prefetches V[src] (4KB, one 128B chunk per lane) into L2 for edge_aggregate.
// ============================================================
__global__ void edge_alpha(const float* __restrict__ Q, const float* __restrict__ K,
                           const float* __restrict__ V,
                           const int* __restrict__ eidx, int E,
                           float* __restrict__ alpha, unsigned* __restrict__ mbuf)
{
    const int wid = (blockIdx.x << 3) + (threadIdx.x >> 5);
    if (wid >= E) return;
    const int lane = threadIdx.x & 31;
    const int src = eidx[wid];
    const int dst = eidx[E + wid];

    // warm L2 for the V gather done later by edge_aggregate (global_prefetch_b8)
    __builtin_prefetch((const char*)(V + (size_t)src * HD) + lane * 128, 0, 0);

    const float4* qv = (const float4*)(Q + (size_t)dst * HD) + lane;
    const float4* kv = (const float4*)(K + (size_t)src * HD) + lane;
    const float scale = 0.08838834764831845f;   // 1/sqrt(128)

    #pragma unroll
    for (int h = 0; h < HEADS; ++h) {
        float4 a = qv[h * 32];
        float4 b = kv[h * 32];
        float s = a.x * b.x + a.y * b.y + a.z * b.z + a.w * b.w;
        #pragma unroll
        for (int off = 16; off >= 1; off >>= 1) s += __shfl_xor(s, off, 32);
        if (lane == 0) {
            const float al = s * scale;
            alpha[(size_t)wid * HEADS + h] = al;
            atomicMax(&mbuf[(size_t)dst * HEADS + h], fenc(al));
        }
    }
}

// ============================================================
// ea = exp(alpha - m[dst]); denom[dst,h] += ea   (thread per (edge,head))
// ============================================================
__global__ void edge_softmax_num(const float* __restrict__ alphaIn, float* __restrict__ ea,
                                 const int* __restrict__ eidx, int E,
                                 const unsigned* __restrict__ mbuf, float* __restrict__ denom)
{
    const int t = blockIdx.x * blockDim.x + threadIdx.x;
    if (t >= E * HEADS) return;
    const int e = t >> 3, h = t & 7;
    const int dst = eidx[E + e];
    const float mm = fdec(mbuf[(size_t)dst * HEADS + h]);
    const float v = __expf(alphaIn[t] - mm);
    ea[t] = v;
    atomicAdd(&denom[(size_t)dst * HEADS + h], v);
}

// ============================================================
// Weighted V scatter with fused head-mean:
// hmean[dst, :] += (ea / denom) * (1/H) * V[src, h, :]   (one wave per (edge,head))
// ============================================================
__global__ void edge_aggregate(const float* __restrict__ V, const float* __restrict__ ea,
                               const float* __restrict__ denom, const int* __restrict__ eidx,
                               int E, float* __restrict__ hmean)
{
    const int gid = (blockIdx.x << 3) + (threadIdx.x >> 5);   // gid = e*8 + h
    if (gid >= E * HEADS) return;
    const int e = gid >> 3, h = gid & 7;
    const int lane = threadIdx.x & 31;
    const int src = eidx[e];
    const int dst = eidx[E + e];

    const float w = ea[gid] / (denom[(size_t)dst * HEADS + h] + 1e-16f) * (1.0f / HEADS);
    const float4 v = *((const float4*)(V + (size_t)src * HD + h * IN_DIM) + lane);
    float* o = hmean + (size_t)dst * IN_DIM + (lane << 2);
    atomicAdd(o + 0, w * v.x);
    atomicAdd(o + 1, w * v.y);
    atomicAdd(o + 2, w * v.z);
    atomicAdd(o + 3, w * v.w);
}

// ============================================================
// Per-graph mean pool (sum stage): gsum[b, d] += h2[n, d]
// ============================================================
__global__ void pool_sum(const float* __restrict__ h2, int N, float* __restrict__ gsum)
{
    const int t = blockIdx.x * blockDim.x + threadIdx.x;
    if (t >= N * OUT_DIM) return;
    const int n = t >> 6, d = t & 63;
    const int b = n / NPG;
    atomicAdd(&gsum[b * OUT_DIM + d], h2[t]);
}

// ============================================================
// Write flattened outputs: [g_cls 16*64][node_tokens 16*128*64][node_mask 16*128]
// ============================================================
__global__ void finalize(const float* __restrict__ gsum, const float* __restrict__ h2,
                         const int* __restrict__ sel, float* __restrict__ out, int total)
{
    const int t = blockIdx.x * blockDim.x + threadIdx.x;
    if (t >= total) return;
    const int GCLS = NGRAPH * OUT_DIM;              // 1024
    const int TOK  = NGRAPH * NTOK * OUT_DIM;       // 131072
    if (t < GCLS) {
        out[t] = gsum[t] * (1.0f / NPG);
    } else if (t < GCLS + TOK) {
        const int r   = t - GCLS;
        const int b   = r / (NTOK * OUT_DIM);
        const int rem = r - b * (NTOK * OUT_DIM);
        const int tok = rem >> 6, d = rem & 63;
        const int node = sel[b * NTOK + tok];
        out[t] = h2[((size_t)b * NPG + node) * OUT_DIM + d];
    } else {
        out[t] = 1.0f;                               // node_mask = True
    }
}

extern "C" void kernel_launch(void* const* d_in, const int* in_sizes, int n_in,
                              void* d_out, int out_size, void* d_ws, size_t ws_size,
                              hipStream_t stream)
{
    (void)n_in; (void)ws_size;
    const float* x    = (const float*)d_in[0];
    const float* Wq   = (const float*)d_in[1];
    const float* bq   = (const float*)d_in[2];
    const float* Wk   = (const float*)d_in[3];
    const float* bk   = (const float*)d_in[4];
    const float* Wv   = (const float*)d_in[5];
    const float* bv   = (const float*)d_in[6];
    const float* Wsk  = (const float*)d_in[7];
    const float* bsk  = (const float*)d_in[8];
    const float* Wemb = (const float*)d_in[9];
    const float* bemb = (const float*)d_in[10];
    const int*   eidx = (const int*)d_in[11];
    const int*   sel  = (const int*)d_in[13];
    float* out = (float*)d_out;

    const int N = in_sizes[0] / IN_DIM;      // 32768
    const int E = in_sizes[11] / 2;          // 262144

    // ---- workspace layout (floats) ----
    float* ws = (float*)d_ws;
    const size_t NQ = (size_t)N * HD;
    float*    Q     = ws;                                      // N*1024
    float*    Kb    = Q  + NQ;                                 // N*1024
    float*    Vb    = Kb + NQ;                                 // N*1024
    float*    alpha = Vb + NQ;                                 // E*8 (reused as ea)
    unsigned* mbuf  = (unsigned*)(alpha + (size_t)E * HEADS);  // N*8
    float*    denom = (float*)mbuf + (size_t)N * HEADS;        // N*8
    float*    hmean = denom + (size_t)N * HEADS;               // N*128
    float*    gsum  = hmean + (size_t)N * IN_DIM;              // 16*64
    float*    h2    = gsum  + NGRAPH * OUT_DIM;                // N*64

    // zero m/denom/hmean/gsum (contiguous block); enc(0) < enc(any real alpha)
    const size_t zbytes = ((size_t)N * HEADS * 2 + (size_t)N * IN_DIM
                           + NGRAPH * OUT_DIM) * sizeof(float);
    hipMemsetAsync(mbuf, 0, zbytes, stream);

    const dim3 blk(128);
    // Q/K/V GEMMs (fp32 WMMA, 16x64 per wave)
    wmma_gemm128<false><<<dim3(N / 64, HD / 64), blk, 0, stream>>>(x, Wq, bq, nullptr, Q,  HD);
    wmma_gemm128<false><<<dim3(N / 64, HD / 64), blk, 0, stream>>>(x, Wk, bk, nullptr, Kb, HD);
    wmma_gemm128<false><<<dim3(N / 64, HD / 64), blk, 0, stream>>>(x, Wv, bv, nullptr, Vb, HD);

    // edge attention pipeline
    edge_alpha<<<dim3((E + 7) / 8), dim3(256), 0, stream>>>(Q, Kb, Vb, eidx, E, alpha, mbuf);
    edge_softmax_num<<<dim3((E * HEADS + 255) / 256), dim3(256), 0, stream>>>(
        alpha, alpha, eidx, E, mbuf, denom);
    edge_aggregate<<<dim3((E * HEADS + 7) / 8), dim3(256), 0, stream>>>(
        Vb, alpha, denom, eidx, E, hmean);

    // skip GEMM with residual add (in place on hmean), then embedding GEMM
    wmma_gemm128<true><<<dim3(N / 64, IN_DIM / 64), blk, 0, stream>>>(x, Wsk, bsk, hmean, hmean, IN_DIM);
    wmma_gemm128<false><<<dim3(N / 64, OUT_DIM / 64), blk, 0, stream>>>(hmean, Wemb, bemb, nullptr, h2, OUT_DIM);

    // pooling + outputs
    pool_sum<<<dim3((N * OUT_DIM + 255) / 256), dim3(256), 0, stream>>>(h2, N, gsum);
    finalize<<<dim3((out_size + 255) / 256), dim3(256), 0, stream>>>(gsum, h2, sel, out, out_size);
}